// MHA_66666482369036
// MI455X (gfx1250) — compile-verified
//
#include <hip/hip_runtime.h>
#include <hip/hip_bf16.h>

typedef __attribute__((ext_vector_type(2))) float v2f;
typedef __attribute__((ext_vector_type(8))) float v8f;

#define WMMA_F32_16x16x4(a, b, c) \
  __builtin_amdgcn_wmma_f32_16x16x4_f32(false, (a), false, (b), (short)0, (c), false, false)

constexpr int BQ = 8;      // batch
constexpr int NH = 32;     // heads
constexpr int HD = 128;    // head dim
constexpr int DM = 4096;   // model dim
constexpr int N3 = 3 * DM; // qkv dim
constexpr int NP = 32;     // pages
constexpr int PS = 128;    // page size

constexpr int KTP = 132;   // padded K-tile row stride (floats): 4-bank lane skew
constexpr int XSP = DM + 4;// padded x row stride in LDS

// Issue one fully-coalesced async copy of 16 B/lane: global -> LDS.
__device__ __forceinline__ void async_cp_b128(unsigned lds_off, const float* gaddr) {
  asm volatile("global_load_async_to_lds_b128 %0, %1, off"
               :: "v"(lds_off), "v"(gaddr) : "memory");
}
__device__ __forceinline__ unsigned lds_off_of(const void* p) {
  return (unsigned)(uintptr_t)p;   // flat LDS addr: low 32 bits == LDS byte offset
}

// ---------------------------------------------------------------------------
// Skinny GEMM: Y[BQ, N] = X[BQ, 4096] * W[4096, N] + bias[N].
// x staged once per block into padded LDS via async b128 copies; one wave per
// 16-column tile; K-loop of v_wmma_f32_16x16x4_f32 with A from LDS and B from
// global through a 32-bit incremented offset (saddr+voffset addressing).
// Padding rows 8..15 of A read clamped row 0 -> pollute only D rows 8..15,
// which are never stored.
// ---------------------------------------------------------------------------
__global__ __launch_bounds__(256)
void mha_gemm16(const float* __restrict__ X, const float* __restrict__ W,
                const float* __restrict__ bias, float* __restrict__ Y, int N) {
  __shared__ __align__(16) float xs[BQ][XSP];   // 8 x 4100 floats

  const int tid  = threadIdx.x;
  const int lane = tid & 31;
  const int wid  = tid >> 5;

  // ---- stage x (8x4096 f32) into LDS: 8192 float4 chunks over 256 threads ----
  {
    const unsigned xsbase = lds_off_of(&xs[0][0]);
#pragma unroll 8
    for (int c = tid; c < BQ * DM / 4; c += 256) {
      const int row = c >> 10;          // / (DM/4)
      const int cid = c & 1023;
      async_cp_b128(xsbase + (unsigned)(row * XSP + cid * 4) * 4u,
                    X + row * DM + cid * 4);
    }
    asm volatile("s_wait_asynccnt 0x0" ::: "memory");
  }
  __syncthreads();

  const int j0   = (blockIdx.x * 8 + wid) * 16;   // N-tile base
  const int m    = lane & 15;                      // A row (batch, padded)
  const int mrow = (m < BQ) ? m : 0;               // clamp padding rows
  const int koff = (lane >> 4) << 1;               // 0 | 2
  const int col  = j0 + m;

  const float* xrow = &xs[mrow][koff];             // LDS, bank-skewed rows
  unsigned woff = (unsigned)koff * (unsigned)N + (unsigned)col;
  const unsigned wstep = 4u * (unsigned)N;

  v8f c = {0.f, 0.f, 0.f, 0.f, 0.f, 0.f, 0.f, 0.f};
#pragma unroll 8
  for (int k = 0; k < DM; k += 4) {
    if ((k & 127) == 0)                            // prime ~128 rows ahead
      __builtin_prefetch(W + woff + 32u * wstep, 0, 1);
    v2f a = *(const v2f*)(xrow + k);               // ds_load_b64
    v2f bv;
    bv.x = W[woff];
    bv.y = W[woff + (unsigned)N];
    c = WMMA_F32_16x16x4(a, bv, c);
    woff += wstep;
  }

  // D: VGPR i, lanes 0-15 -> row M=i (batches 0..7), col = lane
  if (lane < 16) {
    const float bv = bias[j0 + lane];
#pragma unroll
    for (int i = 0; i < 8; ++i)
      Y[(size_t)i * N + j0 + lane] = c[i] + bv;
  }
}

// ---------------------------------------------------------------------------
// Paged decode attention, flash-decoding across 8 waves (4 pages each).
// K tiles (16x128 f32) stream into double-buffered padded LDS via async b128
// copies one tile ahead (ASYNCcnt pipelining); scores via WMMA f32 16x16x4
// with keys in the A rows and q broadcast to every B column; P.V uses
// coalesced float4 rows of V straight from global.
// ---------------------------------------------------------------------------
__global__ __launch_bounds__(256)
void mha_attn(const float* __restrict__ qkv, const float* __restrict__ Kc,
              const float* __restrict__ Vc, float* __restrict__ Yout) {
  const int bh   = blockIdx.x;
  const int b    = bh >> 5;       // / NH
  const int h    = bh & 31;       // % NH
  const int tid  = threadIdx.x;
  const int lane = tid & 31;
  const int wid  = tid >> 5;      // 0..7
  const int e0   = lane * 4;      // this lane's 4 output dims
  const int koff = (lane >> 4) << 1;

  __shared__ __align__(16) float qs[HD];
  __shared__ __align__(16) float ktile[8][2][16 * KTP];  // 8 waves x 2 bufs x 8.45KB
  __shared__ __align__(16) float ylds[8][HD];
  __shared__ float mlds[8], dlds[8];

  const float scale = 0.08838834764831845f;  // 1/sqrt(128)

  if (tid < HD) qs[tid] = qkv[(size_t)b * N3 + h * HD + tid];
  __syncthreads();

  const unsigned ktbase[2] = { lds_off_of(&ktile[wid][0][0]),
                               lds_off_of(&ktile[wid][1][0]) };

  // stage one 16x128 K-tile: 16 rows, 512 B coalesced per instruction,
  // padded to KTP-float rows in LDS
  auto stage_ktile = [&](const float* src, unsigned base) {
#pragma unroll
    for (int i = 0; i < 16; ++i)
      async_cp_b128(base + (unsigned)(i * KTP + lane * 4) * 4u,
                    src + i * HD + lane * 4);
  };

  constexpr int TILES = (NP / 8) * (PS / 16);   // 4 pages x 8 tiles = 32
  auto ktile_src = [&](int it) -> const float* {
    const int p  = wid + (it >> 3) * 8;
    const int t0 = (it & 7) * 16;
    return Kc + (size_t)((p * BQ + b) * NH + h) * (PS * HD) + (size_t)t0 * HD;
  };

  float m = -__builtin_inff();
  float d = 0.f;
  float y0 = 0.f, y1 = 0.f, y2 = 0.f, y3 = 0.f;

  stage_ktile(ktile_src(0), ktbase[0]);

  for (int it = 0; it < TILES; ++it) {
    const int cur = it & 1;
    if (it + 1 < TILES) {
      stage_ktile(ktile_src(it + 1), ktbase[cur ^ 1]);
      asm volatile("s_wait_asynccnt 0x10" ::: "memory"); // current 16 loads done
    } else {
      asm volatile("s_wait_asynccnt 0x0" ::: "memory");
    }

    // ---- scores for 16 keys: S = K_tile(16x128) . q, from LDS ----
    v8f cS = {0.f, 0.f, 0.f, 0.f, 0.f, 0.f, 0.f, 0.f};
    const float* Krow = &ktile[wid][cur][(lane & 15) * KTP + koff];
#pragma unroll 8
    for (int k = 0; k < HD; k += 4) {
      v2f a  = *(const v2f*)(Krow + k);          // ds_load_b64, bank-skewed
      v2f bq = *(const v2f*)(&qs[k + koff]);     // uniform broadcast
      cS = WMMA_F32_16x16x4(a, bq, cS);
    }
    // lanes 0-15 hold keys t0+0..7 in cS[0..7]; lanes 16-31 hold t0+8..15
#pragma unroll
    for (int i = 0; i < 8; ++i) cS[i] *= scale;

    float lmax = cS[0];
#pragma unroll
    for (int i = 1; i < 8; ++i) lmax = fmaxf(lmax, cS[i]);
    const float tmax = fmaxf(__shfl(lmax, 0), __shfl(lmax, 16));

    const float mnew = fmaxf(m, tmax);
    const float corr = __expf(m - mnew);
    d *= corr; y0 *= corr; y1 *= corr; y2 *= corr; y3 *= corr;
    m = mnew;

    // ---- P.V with coalesced float4 rows of V ----
    const int p  = wid + (it >> 3) * 8;
    const int t0 = (it & 7) * 16;
    const float* Vt = Vc + (size_t)((p * BQ + b) * NH + h) * (PS * HD)
                         + (size_t)t0 * HD;
#pragma unroll
    for (int j = 0; j < 16; ++j) {
      const float sj = __shfl(cS[j & 7], (j >> 3) * 16);  // key t0+j, all lanes
      const float pj = __expf(sj - mnew);
      d += pj;
      const float4 vv = *(const float4*)(Vt + (size_t)j * HD + e0);
      y0 += pj * vv.x; y1 += pj * vv.y; y2 += pj * vv.z; y3 += pj * vv.w;
    }
  }

  // wave 0 folds in the freshly computed k/v token
  if (wid == 0) {
    const float* kn = qkv + (size_t)b * N3 + DM     + h * HD;
    const float* vn = qkv + (size_t)b * N3 + 2 * DM + h * HD;
    float part = 0.f;
#pragma unroll
    for (int i = 0; i < 4; ++i) part += qs[e0 + i] * kn[e0 + i];
#pragma unroll
    for (int off = 16; off >= 1; off >>= 1) part += __shfl_xor(part, off);
    const float s    = part * scale;
    const float mnew = fmaxf(m, s);
    const float corr = __expf(m - mnew);
    const float pj   = __expf(s - mnew);
    const float4 vv  = *(const float4*)(vn + e0);
    d  = d * corr + pj;
    y0 = y0 * corr + pj * vv.x;
    y1 = y1 * corr + pj * vv.y;
    y2 = y2 * corr + pj * vv.z;
    y3 = y3 * corr + pj * vv.w;
    m  = mnew;
  }

  // cross-wave combine via LDS
  ylds[wid][e0 + 0] = y0; ylds[wid][e0 + 1] = y1;
  ylds[wid][e0 + 2] = y2; ylds[wid][e0 + 3] = y3;
  if (lane == 0) { mlds[wid] = m; dlds[wid] = d; }
  __syncthreads();

  if (tid < HD) {
    float ms = mlds[0];
#pragma unroll
    for (int w = 1; w < 8; ++w) ms = fmaxf(ms, mlds[w]);
    float den = 0.f, num = 0.f;
#pragma unroll
    for (int w = 0; w < 8; ++w) {
      const float f = __expf(mlds[w] - ms);
      den += f * dlds[w];
      num += f * ylds[w][tid];
    }
    Yout[(size_t)bh * HD + tid] = num / den;
  }
}

// ---------------------------------------------------------------------------
extern "C" void kernel_launch(void* const* d_in, const int* in_sizes, int n_in,
                              void* d_out, int out_size, void* d_ws, size_t ws_size,
                              hipStream_t stream) {
  const float* x   = (const float*)d_in[0];
  const float* kc  = (const float*)d_in[1];
  const float* vc  = (const float*)d_in[2];
  const float* caw = (const float*)d_in[3];
  const float* cab = (const float*)d_in[4];
  const float* cpw = (const float*)d_in[5];
  const float* cpb = (const float*)d_in[6];
  float* out = (float*)d_out;

  float* qkv = (float*)d_ws;              // [8, 12288] fp32
  float* yat = qkv + (size_t)BQ * N3;     // [8, 32*128] == [8, 4096] fp32

  // 1) qkv = x @ c_attn_w + c_attn_b   (768 N-tiles, 8 waves/block)
  mha_gemm16<<<dim3(N3 / 16 / 8), 256, 0, stream>>>(x, caw, cab, qkv, N3);
  // 2) paged online-softmax attention  (one block per (b,h), 8 waves)
  mha_attn<<<dim3(BQ * NH), 256, 0, stream>>>(qkv, kc, vc, yat);
  // 3) out = y @ c_proj_w + c_proj_b   (256 N-tiles, 8 waves/block)
  mha_gemm16<<<dim3(DM / 16 / 8), 256, 0, stream>>>(yat, cpw, cpb, out, DM);
}